// RandomizedNodeLabeling_73186242724539
// MI455X (gfx1250) — compile-verified
//
#include <hip/hip_runtime.h>

typedef __attribute__((ext_vector_type(2))) float v2f;
typedef __attribute__((ext_vector_type(8))) float v8f;

#define DIM 64

// -------------------------------------------------------------------------
// zero scratch (float4 granularity)
// -------------------------------------------------------------------------
__global__ void rnl_zero_kernel(float4* __restrict__ p, long n4) {
  long i = (long)blockIdx.x * blockDim.x + threadIdx.x;
  if (i < n4) p[i] = make_float4(0.f, 0.f, 0.f, 0.f);
}

// -------------------------------------------------------------------------
// xw = x * node_weight[:, None]   (float4 per thread, 16 float4 per node)
// -------------------------------------------------------------------------
__global__ void rnl_scale_kernel(const float4* __restrict__ x,
                                 const float* __restrict__ w,
                                 float4* __restrict__ xw, long n4) {
  long i = (long)blockIdx.x * blockDim.x + threadIdx.x;
  if (i >= n4) return;
  const long node = i >> 4;  // 64 floats = 16 float4 per node
  const float s = w[node];
  float4 v = x[i];
  v.x *= s; v.y *= s; v.z *= s; v.w *= s;
  xw[i] = v;
}

// -------------------------------------------------------------------------
// Segment-sum SpMM over sorted COO:
//   dst[row] += src[col]  (64 dims, 2 per lane)
//   sc_dst[row] += (sc_src ? sc_src[col] : 1.0)
// One wave per 128-edge chunk; register accumulation across row runs,
// atomic flush only at run boundaries (adj_row is sorted).
// -------------------------------------------------------------------------
__global__ void rnl_spmm_kernel(const float* __restrict__ src,
                                const int* __restrict__ adjr,
                                const int* __restrict__ adjc,
                                int nE,
                                float* __restrict__ dst,
                                const float* __restrict__ sc_src,
                                float* __restrict__ sc_dst) {
  const int lane = threadIdx.x & 31;
  const int wave = ((int)blockIdx.x * (int)blockDim.x + (int)threadIdx.x) >> 5;
  const int base = wave * 128;
  if (base >= nE) return;

  int cur = -1;
  float ax = 0.f, ay = 0.f, s = 0.f;

  for (int b = 0; b < 128; b += 32) {
    const int e = base + b + lane;
    // prefetch the next chunk of adjacency indices (speculative, gfx1250)
    __builtin_prefetch(adjr + base + b + 128, 0, 0);
    __builtin_prefetch(adjc + base + b + 128, 0, 0);
    const int r = (e < nE) ? adjr[e] : -1;
    const int c = (e < nE) ? adjc[e] : 0;
    for (int j = 0; j < 32; ++j) {
      const int row = __shfl(r, j, 32);
      if (row < 0) break;  // wave-uniform tail
      const int col = __shfl(c, j, 32);
      if (row != cur) {
        if (cur >= 0) {
          float* p = dst + (size_t)cur * DIM + 2 * lane;
          atomicAdd(p, ax);
          atomicAdd(p + 1, ay);
          if (lane == 0) atomicAdd(sc_dst + cur, s);
        }
        cur = row;
        ax = 0.f; ay = 0.f; s = 0.f;
      }
      const float2 v = *(const float2*)(src + (size_t)col * DIM + 2 * lane);
      ax += v.x;
      ay += v.y;
      s += sc_src ? sc_src[col] : 1.0f;
    }
  }
  if (cur >= 0) {
    float* p = dst + (size_t)cur * DIM + 2 * lane;
    atomicAdd(p, ax);
    atomicAdd(p + 1, ay);
    if (lane == 0) atomicAdd(sc_dst + cur, s);
  }
}

// -------------------------------------------------------------------------
// diag extraction helper for 16x16 f32 C/D tile
// lanes 0-7 own edges 0-7 (element c[lane]); lanes 24-31 own edges 8-15
// (element c[lane-24]).
// -------------------------------------------------------------------------
__device__ __forceinline__ float rnl_pick(v8f c, int ci) {
  float r = c[0];
  if (ci == 1) r = c[1];
  if (ci == 2) r = c[2];
  if (ci == 3) r = c[3];
  if (ci == 4) r = c[4];
  if (ci == 5) r = c[5];
  if (ci == 6) r = c[6];
  if (ci == 7) r = c[7];
  return r;
}

// -------------------------------------------------------------------------
// Per-query-edge features. One wave handles 16 edges; the 9 distinct
// 64-dim dot products are computed as the diagonals of 16x16 WMMA tiles,
// chained over K with V_WMMA_F32_16X16X4_F32 (full f32 precision).
//
// Fragment layout (ISA 7.12.2, 32-bit A 16x4 and B 4x16): for K-block kb,
// lane L supplies a float2 from the row vector of edge (L & 15) at offset
// 4*kb + 2*(L >> 4) -- identical addressing for A and B fragments.
// -------------------------------------------------------------------------
__global__ void rnl_feat_kernel(const float* __restrict__ xw,
                                const float* __restrict__ oh,
                                const float* __restrict__ it,
                                const float* __restrict__ deg,
                                const float* __restrict__ degs,
                                const int* __restrict__ erow,
                                const int* __restrict__ ecol,
                                float* __restrict__ out, int nQ) {
  const int lane = threadIdx.x & 31;
  const int wave = ((int)blockIdx.x * (int)blockDim.x + (int)threadIdx.x) >> 5;
  const long base = (long)wave * 16;
  const int ei = lane & 15;
  const int half = lane >> 4;

  long ec = base + ei;
  if (ec > (long)nQ - 1) ec = (long)nQ - 1;  // clamp: keeps EXEC all-1s
  const int u = erow[ec];
  const int v = ecol[ec];
  const float du = deg[u];
  const float dv = deg[v];

  const float* pxu = xw + (size_t)u * DIM;
  const float* pxv = xw + (size_t)v * DIM;
  const float* pou = oh + (size_t)u * DIM;
  const float* pov = oh + (size_t)v * DIM;
  const float* piu = it + (size_t)u * DIM;
  const float* piv = it + (size_t)v * DIM;

  v8f a1 = {}, a2 = {}, a3 = {}, a4 = {}, a5 = {};
  v8f a6 = {}, a7 = {}, a8 = {}, a9 = {};

#pragma unroll 4
  for (int kb = 0; kb < 16; ++kb) {
    const int ofs = kb * 4 + half * 2;
    const v2f xu = *(const v2f*)(pxu + ofs);
    const v2f xv = *(const v2f*)(pxv + ofs);
    const v2f ou = *(const v2f*)(pou + ofs);
    const v2f ov = *(const v2f*)(pov + ofs);
    const v2f iu = *(const v2f*)(piu + ofs);
    const v2f iv = *(const v2f*)(piv + ofs);
    const v2f tu = iu - ou - xu;          // two_hop(u)
    const v2f tv = iv - ov - xv;          // two_hop(v)
    const v2f gu = iu - du * xu;          // two_iter(u) - deg_u * xw(u)
    const v2f gv = iv - dv * xv;

    a1 = __builtin_amdgcn_wmma_f32_16x16x4_f32(false, ou, false, ov, (short)0, a1, false, false);
    a2 = __builtin_amdgcn_wmma_f32_16x16x4_f32(false, ou, false, tv, (short)0, a2, false, false);
    a3 = __builtin_amdgcn_wmma_f32_16x16x4_f32(false, tu, false, ov, (short)0, a3, false, false);
    a4 = __builtin_amdgcn_wmma_f32_16x16x4_f32(false, tu, false, tv, (short)0, a4, false, false);
    a5 = __builtin_amdgcn_wmma_f32_16x16x4_f32(false, ou, false, iv, (short)0, a5, false, false);
    a6 = __builtin_amdgcn_wmma_f32_16x16x4_f32(false, iu, false, ov, (short)0, a6, false, false);
    a7 = __builtin_amdgcn_wmma_f32_16x16x4_f32(false, gu, false, gv, (short)0, a7, false, false);
    a8 = __builtin_amdgcn_wmma_f32_16x16x4_f32(false, ou, false, iu, (short)0, a8, false, false);
    a9 = __builtin_amdgcn_wmma_f32_16x16x4_f32(false, ov, false, iv, (short)0, a9, false, false);
  }

  const bool low = lane < 8;
  const bool high = lane >= 24;
  const int ci = low ? lane : (lane - 24);

  const float q1 = rnl_pick(a1, ci);
  const float q2 = rnl_pick(a2, ci);
  const float q3 = rnl_pick(a3, ci);
  const float q4 = rnl_pick(a4, ci);
  const float q5 = rnl_pick(a5, ci);
  const float q6 = rnl_pick(a6, ci);
  const float q7 = rnl_pick(a7, ci);
  const float q8 = rnl_pick(a8, ci);
  const float q9 = rnl_pick(a9, ci);

  if (low || high) {
    const long e = base + (low ? lane : (lane - 16));
    if (e < nQ) {
      const float d2u = fmaxf(degs[u] - du - 1.0f, 0.0f);
      const float d2v = fmaxf(degs[v] - dv - 1.0f, 0.0f);
      float* o = out + e * 15;
      o[0]  = q1;                              // count_1_1
      o[1]  = q2;                              // count_1_2
      o[2]  = q3;                              // count_2_1
      o[3]  = q4;                              // count_2_2
      o[4]  = du + dv - 2.f * q1 - q2 - q3;    // count_1_inf
      o[5]  = d2u + d2v - 2.f * q4 - q2 - q3;  // count_2_inf
      o[6]  = q5;                              // comb_count_1_2
      o[7]  = q6;                              // comb_count_2_1
      o[8]  = q7;                              // comb_count_2_2
      o[9]  = q8;                              // comb_count_self_1_2
      o[10] = q9;                              // comb_count_self_2_1
      o[11] = du;
      o[12] = dv;
      o[13] = d2u;
      o[14] = d2v;
    }
  }
}

// -------------------------------------------------------------------------
// host launcher
// -------------------------------------------------------------------------
extern "C" void kernel_launch(void* const* d_in, const int* in_sizes, int n_in,
                              void* d_out, int out_size, void* d_ws, size_t ws_size,
                              hipStream_t stream) {
  const float* x = (const float*)d_in[0];      // [N, 64]
  const float* w = (const float*)d_in[1];      // [N]
  const int* edges = (const int*)d_in[2];      // [2, nQ]
  const int* adjr = (const int*)d_in[3];       // [nE] sorted
  const int* adjc = (const int*)d_in[4];       // [nE]

  const int N = in_sizes[1];
  const int nQ = in_sizes[2] / 2;
  const int nE = in_sizes[3];

  const size_t NF = (size_t)N * DIM;
  const size_t Npad = ((size_t)N + 255) & ~(size_t)255;

  float* xw = (float*)d_ws;        // [N*64]
  float* oh = xw + NF;             // one_hop  [N*64]
  float* it = oh + NF;             // two_iter [N*64]
  float* dg = it + NF;             // deg      [Npad]
  float* dgs = dg + Npad;          // degs     [Npad]

  // zero one_hop / two_iter / deg / degs (contiguous region)
  {
    const long n4 = (long)(2 * NF + 2 * Npad) / 4;
    rnl_zero_kernel<<<(int)((n4 + 255) / 256), 256, 0, stream>>>((float4*)oh, n4);
  }
  // xw = x * w
  {
    const long n4 = (long)(NF / 4);
    rnl_scale_kernel<<<(int)((n4 + 255) / 256), 256, 0, stream>>>(
        (const float4*)x, w, (float4*)xw, n4);
  }
  // SpMM passes
  {
    const int waves = (nE + 127) / 128;
    const int blocks = (waves * 32 + 255) / 256;
    rnl_spmm_kernel<<<blocks, 256, 0, stream>>>(xw, adjr, adjc, nE, oh, nullptr, dg);
    rnl_spmm_kernel<<<blocks, 256, 0, stream>>>(oh, adjr, adjc, nE, it, dg, dgs);
  }
  // features
  {
    const int waves = (nQ + 15) / 16;
    const int blocks = (waves * 32 + 255) / 256;
    rnl_feat_kernel<<<blocks, 256, 0, stream>>>(xw, oh, it, dg, dgs,
                                                edges, edges + nQ,
                                                (float*)d_out, nQ);
  }
}